// QLSTM_65481071408405
// MI455X (gfx1250) — compile-verified
//
#include <hip/hip_runtime.h>
#include <cstddef>

// ---------------------------------------------------------------------------
// LSTM forward for MI455X (gfx1250, wave32, WMMA).
//
//   T=512, B=64, D=512, H=512.  outputs[T,B,H] then hx[B,H], cx[B,H] in d_out.
//
// Strategy:
//   1) lstm_xgemm: Gx[t,b,:] = X[t,b,:] @ W[0:D,:] + bias   (one big fp32
//      WMMA GEMM, 32768x512x2048, fully parallel; latency hidden by
//      occupancy: ~7 blocks / 320KB WGP).
//   2) 512x lstm_step (serial critical path): gates = Gx[t] + h_{t-1} @
//      W[D:D+H,:] with a fused sigmoid/tanh cell update.  Each wave owns a
//      16x16 (batch x hid) patch of ALL FOUR gates -> register-local cell
//      epilogue.  Since only ~1 block/WGP runs, the K-loop is software-
//      pipelined with double-buffered LDS so L2 load latency overlaps WMMA
//      issue.  W (8MB) stays L2-resident across all 512 steps.
// Precision: fp32 via V_WMMA_F32_16X16X4_F32 (matches fp32 reference).
// Workspace: T*B*4H*4 bytes (256MB) for Gx + B*H*4 for c-state.
// ---------------------------------------------------------------------------

#define T_STEPS 512
#define BATCH   64
#define DIM     512
#define HID     512
#define G4      2048   // 4*H
#define BM      64
#define BN      256
#define BK      32
#define SBM     32     // batch rows per lstm_step block

typedef __attribute__((ext_vector_type(2))) float v2f;
typedef __attribute__((ext_vector_type(8))) float v8f;

__device__ __forceinline__ v8f wmma_f32(v2f a, v2f b, v8f c) {
  // D = A(16x4) x B(4x16) + C, fp32 tensor path.
  return __builtin_amdgcn_wmma_f32_16x16x4_f32(
      /*neg_a=*/false, a, /*neg_b=*/false, b,
      /*c_mod=*/(short)0, c, /*reuse_a=*/false, /*reuse_b=*/false);
}

__device__ __forceinline__ float sigf(float x) {
  return 1.0f / (1.0f + __expf(-x));
}

// ---------------------------------------------------------------------------
// Kernel A: Gx = X @ W[0:D,:] + bias.   M=T*B=32768, K=512, N=2048.
// Block = 64x256, K-tile 32, 8 waves; wave tile = 32x64 (2x4 WMMA tiles).
// ---------------------------------------------------------------------------
__global__ __launch_bounds__(256) void lstm_xgemm(
    const float* __restrict__ X, const float* __restrict__ W,
    const float* __restrict__ bias, float* __restrict__ Gx) {
  __shared__ float As[BM][BK + 4];    // 64 x 36  (pad kills bank conflicts)
  __shared__ float Bs[BK][BN + 4];    // 32 x 260

  const int tid  = threadIdx.x;
  const int lane = tid & 31;
  const int wave = tid >> 5;          // 0..7
  const int l16  = lane & 15;
  const int half = lane >> 4;         // 0..1
  const int wm   = wave >> 2;         // 0..1 : 32-row strip
  const int wn   = wave & 3;          // 0..3 : 64-col strip
  const int bm0  = blockIdx.y * BM;
  const int bn0  = blockIdx.x * BN;

  v8f acc[2][4];
#pragma unroll
  for (int i = 0; i < 2; ++i)
#pragma unroll
    for (int j = 0; j < 4; ++j)
#pragma unroll
      for (int e = 0; e < 8; ++e) acc[i][j][e] = 0.0f;

  for (int k0 = 0; k0 < DIM; k0 += BK) {
    // Stage A tile (64x32) : 512 float4 over 256 threads.
#pragma unroll
    for (int i = 0; i < 2; ++i) {
      int idx = tid + i * 256;
      int r = idx >> 3, c = (idx & 7) << 2;
      *(float4*)&As[r][c] =
          *(const float4*)&X[(size_t)(bm0 + r) * DIM + k0 + c];
    }
    // Stage B tile (32x256) : 2048 float4 over 256 threads.
#pragma unroll
    for (int i = 0; i < 8; ++i) {
      int idx = tid + i * 256;
      int r = idx >> 6, c = (idx & 63) << 2;
      *(float4*)&Bs[r][c] =
          *(const float4*)&W[(size_t)(k0 + r) * G4 + bn0 + c];
    }
    if (k0 + BK < DIM)  // hint next W tile into cache (global_prefetch_b8)
      __builtin_prefetch(&W[(size_t)(k0 + BK) * G4 + bn0 + ((tid & 63) << 2)],
                         0, 0);
    __syncthreads();

#pragma unroll
    for (int kk = 0; kk < BK; kk += 4) {
      v2f a[2], b[4];
#pragma unroll
      for (int i = 0; i < 2; ++i)
        a[i] = *(const v2f*)&As[wm * 32 + i * 16 + l16][kk + 2 * half];
#pragma unroll
      for (int j = 0; j < 4; ++j) {
        int col = wn * 64 + j * 16 + l16;
        b[j].x = Bs[kk + 2 * half][col];
        b[j].y = Bs[kk + 2 * half + 1][col];
      }
#pragma unroll
      for (int i = 0; i < 2; ++i)
#pragma unroll
        for (int j = 0; j < 4; ++j) acc[i][j] = wmma_f32(a[i], b[j], acc[i][j]);
    }
    __syncthreads();
  }

  // Store C (+bias).  C layout: VGPR r -> row r + 8*half, col = l16.
#pragma unroll
  for (int i = 0; i < 2; ++i) {
    int row_base = bm0 + wm * 32 + i * 16 + 8 * half;
#pragma unroll
    for (int j = 0; j < 4; ++j) {
      int col = bn0 + wn * 64 + j * 16 + l16;
      float bb = bias[col];
#pragma unroll
      for (int r = 0; r < 8; ++r)
        Gx[(size_t)(row_base + r) * G4 + col] = acc[i][j][r] + bb;
    }
  }
}

// ---------------------------------------------------------------------------
// Kernel B: one recurrent step (critical path).
//   gates = Gx[t] + h_{t-1} @ W[D:D+H,:];  fused cell update.
// Grid = (H/16, B/32) = (32, 2) blocks, 2 waves/block.
// Each wave holds 4 gate accumulators for its 16x16 (batch x hid) patch.
// K-loop double-buffered: tile k+1 global loads overlap tile k WMMA.
// ---------------------------------------------------------------------------
__global__ __launch_bounds__(64) void lstm_step(
    const float* __restrict__ W, const float* __restrict__ Gx,
    float* __restrict__ out, float* __restrict__ cbuf, int t) {
  __shared__ float Hs[2][SBM][36];  // 2 x 32 x (32+4): h_{t-1} K-tiles
  __shared__ float Ws[2][32][68];   // 2 x 32 x (4 gates x 16 + 4): W K-tiles

  const int tid  = threadIdx.x;
  const int lane = tid & 31;
  const int wave = tid >> 5;       // 0..1 : 16-row batch strip
  const int l16  = lane & 15;
  const int half = lane >> 4;
  const int m0   = wave * 16;                // within block
  const int bb0  = blockIdx.y * SBM;         // batch base
  const int h0   = blockIdx.x * 16;          // hidden-column base

  v8f acc[4];
#pragma unroll
  for (int g = 0; g < 4; ++g)
#pragma unroll
    for (int e = 0; e < 8; ++e) acc[g][e] = 0.0f;

  if (t > 0) {
    const float* hprev = out + (size_t)(t - 1) * BATCH * HID;
    float4 hreg[4];   // 32x32 h tile   : 256 float4 / 64 thr
    float4 wreg[8];   // 32x64 W tile   : 512 float4 / 64 thr

    auto loadRegs = [&](int k0) {
#pragma unroll
      for (int i = 0; i < 4; ++i) {
        int idx = tid + i * 64;
        int r = idx >> 3, c = (idx & 7) << 2;
        hreg[i] = *(const float4*)&hprev[(size_t)(bb0 + r) * HID + k0 + c];
      }
#pragma unroll
      for (int i = 0; i < 8; ++i) {
        int idx = tid + i * 64;
        int r = idx >> 4;                    // 0..31
        int cq = idx & 15;
        int g = cq >> 2, c4 = (cq & 3) << 2;
        wreg[i] = *(const float4*)
            &W[(size_t)(DIM + k0 + r) * G4 + g * 512 + h0 + c4];
      }
    };
    auto storeLDS = [&](int buf) {
#pragma unroll
      for (int i = 0; i < 4; ++i) {
        int idx = tid + i * 64;
        int r = idx >> 3, c = (idx & 7) << 2;
        *(float4*)&Hs[buf][r][c] = hreg[i];
      }
#pragma unroll
      for (int i = 0; i < 8; ++i) {
        int idx = tid + i * 64;
        int r = idx >> 4;
        int cq = idx & 15;
        int g = cq >> 2, c4 = (cq & 3) << 2;
        *(float4*)&Ws[buf][r][g * 16 + c4] = wreg[i];
      }
    };

    loadRegs(0);
    storeLDS(0);
    __syncthreads();

    const int NT = HID / 32;     // 16 K-tiles
    for (int kt = 0; kt < NT; ++kt) {
      if (kt + 1 < NT) loadRegs((kt + 1) * 32);  // overlap with compute
      const int buf = kt & 1;
#pragma unroll
      for (int kk = 0; kk < 32; kk += 4) {
        v2f a = *(const v2f*)&Hs[buf][m0 + l16][kk + 2 * half];
#pragma unroll
        for (int g = 0; g < 4; ++g) {
          v2f b;
          b.x = Ws[buf][kk + 2 * half][g * 16 + l16];
          b.y = Ws[buf][kk + 2 * half + 1][g * 16 + l16];
          acc[g] = wmma_f32(a, b, acc[g]);
        }
      }
      if (kt + 1 < NT) {
        __syncthreads();              // everyone done reading buf^1 (iter kt-1)
        storeLDS(buf ^ 1);
        __syncthreads();              // buf^1 ready for iter kt+1
      }
    }
  }

  // Fused LSTM cell epilogue: gate columns [f,i,g,o] are 512 apart in Gx.
  const float* gx = Gx + (size_t)t * BATCH * G4;
  float* ht = out + (size_t)t * BATCH * HID;
#pragma unroll
  for (int r = 0; r < 8; ++r) {
    int brow = bb0 + m0 + r + 8 * half;
    int hcol = h0 + l16;
    size_t gb = (size_t)brow * G4 + hcol;
    float fg = sigf(acc[0][r] + gx[gb]);
    float ig = sigf(acc[1][r] + gx[gb + 512]);
    float gg = tanhf(acc[2][r] + gx[gb + 1024]);
    float og = sigf(acc[3][r] + gx[gb + 1536]);
    size_t bh = (size_t)brow * HID + hcol;
    float cp = (t > 0) ? cbuf[bh] : 0.0f;
    float cn = fg * cp + ig * gg;
    ht[bh] = og * tanhf(cn);
    cbuf[bh] = cn;
  }
}

// ---------------------------------------------------------------------------
extern "C" void kernel_launch(void* const* d_in, const int* in_sizes, int n_in,
                              void* d_out, int out_size, void* d_ws,
                              size_t ws_size, hipStream_t stream) {
  (void)in_sizes; (void)n_in; (void)out_size; (void)ws_size;
  const float* X    = (const float*)d_in[0];  // [T,B,D]
  const float* W    = (const float*)d_in[1];  // [D+H, 4H]
  const float* bias = (const float*)d_in[2];  // [4H]
  float* out  = (float*)d_out;                // [T*B*H + B*H + B*H]
  float* Gx   = (float*)d_ws;                                   // 256 MB
  float* cbuf = Gx + (size_t)T_STEPS * BATCH * G4;              // 128 KB

  // 1) Input-projection GEMM for all timesteps (parallel).
  dim3 gA(G4 / BN, (T_STEPS * BATCH) / BM);   // (8, 512)
  lstm_xgemm<<<gA, 256, 0, stream>>>(X, W, bias, Gx);

  // 2) Sequential recurrence: 512 graph nodes, stream-ordered.
  for (int t = 0; t < T_STEPS; ++t)
    lstm_step<<<dim3(HID / 16, BATCH / SBM), 64, 0, stream>>>(W, Gx, out,
                                                              cbuf, t);

  // 3) hx = outputs[T-1], cx = final c-state.
  const size_t BH = (size_t)BATCH * HID;
  hipMemcpyAsync(out + (size_t)T_STEPS * BH, out + (size_t)(T_STEPS - 1) * BH,
                 BH * sizeof(float), hipMemcpyDeviceToDevice, stream);
  hipMemcpyAsync(out + (size_t)T_STEPS * BH + BH, cbuf, BH * sizeof(float),
                 hipMemcpyDeviceToDevice, stream);
}